// PatientGNNSAGE_19172734009898
// MI455X (gfx1250) — compile-verified
//
#include <hip/hip_runtime.h>

#define N_PATIENT 50000
#define N_GENE    20000
#define N_EDGE    640000
#define P_DIM 64
#define G_DIM 32
#define H     128
#define OUT_D 8

#define SCAN_THREADS 1024
#define SCAN_CHUNK   ((N_PATIENT + SCAN_THREADS - 1) / SCAN_THREADS)   // 49

typedef __attribute__((ext_vector_type(2))) float v2f;
typedef __attribute__((ext_vector_type(8))) float v8f;

// ---------------------------------------------------------------------------
// Generic GEMM: C[M,128] = act( A[M,K] @ W[K,128] (+ A2[M,128] @ W2[128,128]) + bias )
// One 256-thread block (8 wave32) computes a 64x128 output tile.
// Wave (w&3) -> 16-row M sub-tile, wave (w>>2) -> 64-col N half.
// Each wave: 4 accumulators of V_WMMA_F32_16X16X4_F32.
// ---------------------------------------------------------------------------
template<bool RELU, bool DUAL>
__global__ __launch_bounds__(256) void gemm_h128(
    const float* __restrict__ A, int K,
    const float* __restrict__ W,
    const float* __restrict__ A2,
    const float* __restrict__ W2,
    const float* __restrict__ bias,
    float* __restrict__ C, int M)
{
  const int lane = threadIdx.x & 31;
  const int wave = threadIdx.x >> 5;
  const int half = lane >> 4;   // 0: lanes 0-15, 1: lanes 16-31
  const int l16  = lane & 15;
  const int m_base = blockIdx.x * 64 + (wave & 3) * 16;
  const int n_base = (wave >> 2) * 64;

  // A-fragment row for this lane (clamped so tail blocks read valid memory;
  // garbage rows are discarded by the predicated store).
  int arow = m_base + l16;
  if (arow > M - 1) arow = M - 1;

  v8f zero = {0.f, 0.f, 0.f, 0.f, 0.f, 0.f, 0.f, 0.f};
  v8f acc[4];
  acc[0] = zero; acc[1] = zero; acc[2] = zero; acc[3] = zero;

  auto mma_pair = [&](const float* __restrict__ Ap,
                      const float* __restrict__ Wp, int Kp) {
    const float* arp = Ap + (size_t)arow * Kp;
    for (int k = 0; k < Kp; k += 4) {
      // 32-bit A layout (16x4): lanes 0-15 -> K = {k, k+1}; lanes 16-31 -> {k+2, k+3}
      v2f a = *(const v2f*)(arp + k + 2 * half);
#pragma unroll
      for (int j = 0; j < 4; ++j) {
        const int ncol = n_base + j * 16 + l16;
        // B layout (4x16): VGPR0 = rows {k, k+2}, VGPR1 = rows {k+1, k+3}
        v2f b;
        b.x = Wp[(size_t)(k + 2 * half + 0) * H + ncol];
        b.y = Wp[(size_t)(k + 2 * half + 1) * H + ncol];
        acc[j] = __builtin_amdgcn_wmma_f32_16x16x4_f32(
            false, a, false, b, (short)0, acc[j], false, false);
      }
    }
  };

  mma_pair(A, W, K);
  if (DUAL) mma_pair(A2, W2, H);

  // C/D layout: VGPR r, lanes 0-15 -> row m_base+r, lanes 16-31 -> row m_base+8+r
  const int rstore = m_base + half * 8;
#pragma unroll
  for (int j = 0; j < 4; ++j) {
    const int ncol = n_base + j * 16 + l16;
    const float bv = bias[ncol];
#pragma unroll
    for (int r = 0; r < 8; ++r) {
      const int row = rstore + r;
      if (row < M) {
        float v = acc[j][r] + bv;
        if (RELU) v = v > 0.f ? v : 0.f;
        C[(size_t)row * H + ncol] = v;
      }
    }
  }
}

// ---------------------------------------------------------------------------
// CSR build step 1: in-degree histogram over dst.
// ---------------------------------------------------------------------------
__global__ __launch_bounds__(256) void edge_count(
    const int* __restrict__ edges, int* __restrict__ deg)
{
  const int e = blockIdx.x * 256 + threadIdx.x;
  if (e >= N_EDGE) return;
  atomicAdd(&deg[edges[N_EDGE + e]], 1);
}

// ---------------------------------------------------------------------------
// CSR build step 2: single-block exclusive scan of deg[0..N_PATIENT) into
// offs[] (and a working copy cursor[]). Per-thread chunk sums + LDS
// Hillis-Steele scan over the 1024 partials.
// ---------------------------------------------------------------------------
__global__ __launch_bounds__(SCAN_THREADS) void deg_scan(
    const int* __restrict__ deg, int* __restrict__ offs, int* __restrict__ cursor)
{
  __shared__ int sdata[SCAN_THREADS];
  const int t = threadIdx.x;
  const int beg = t * SCAN_CHUNK;
  int end = beg + SCAN_CHUNK;
  if (end > N_PATIENT) end = N_PATIENT;

  int mySum = 0;
  for (int i = beg; i < end; ++i) mySum += deg[i];
  sdata[t] = mySum;
  __syncthreads();

  // inclusive scan of partials
  for (int off = 1; off < SCAN_THREADS; off <<= 1) {
    int v = (t >= off) ? sdata[t - off] : 0;
    __syncthreads();
    sdata[t] += v;
    __syncthreads();
  }

  int running = sdata[t] - mySum;   // exclusive base for this chunk
  for (int i = beg; i < end; ++i) {
    offs[i] = running;
    cursor[i] = running;
    running += deg[i];
  }
  if (t == SCAN_THREADS - 1) offs[N_PATIENT] = sdata[SCAN_THREADS - 1];
}

// ---------------------------------------------------------------------------
// CSR build step 3: bucket the source ids by destination.
// ---------------------------------------------------------------------------
__global__ __launch_bounds__(256) void edge_fill(
    const int* __restrict__ edges, int* __restrict__ cursor,
    int* __restrict__ src_sorted)
{
  const int e = blockIdx.x * 256 + threadIdx.x;
  if (e >= N_EDGE) return;
  const int dst = edges[N_EDGE + e];
  const int pos = atomicAdd(&cursor[dst], 1);
  src_sorted[pos] = edges[e];
}

// ---------------------------------------------------------------------------
// Aggregation as a gather: one wave per patient; lane owns 4 columns.
// agg[p] = sum_{i in [offs[p],offs[p+1])} g[src_sorted[i]]
// Fully writes agg (rows with no edges get zeros) -> no memset needed.
// ---------------------------------------------------------------------------
__global__ __launch_bounds__(256) void gather_agg(
    const float* __restrict__ g,
    const int* __restrict__ offs,
    const int* __restrict__ src_sorted,
    float* __restrict__ agg)
{
  const int p = blockIdx.x * 8 + (threadIdx.x >> 5);
  if (p >= N_PATIENT) return;
  const int lane = threadIdx.x & 31;
  const int beg = offs[p];
  const int end = offs[p + 1];

  float4 acc0 = {0.f, 0.f, 0.f, 0.f};
  float4 acc1 = {0.f, 0.f, 0.f, 0.f};
  int i = beg;
  for (; i + 1 < end; i += 2) {
    const int s0 = src_sorted[i];
    const int s1 = src_sorted[i + 1];
    const float4 v0 = *(const float4*)(g + (size_t)s0 * H + lane * 4);
    const float4 v1 = *(const float4*)(g + (size_t)s1 * H + lane * 4);
    acc0.x += v0.x; acc0.y += v0.y; acc0.z += v0.z; acc0.w += v0.w;
    acc1.x += v1.x; acc1.y += v1.y; acc1.z += v1.z; acc1.w += v1.w;
  }
  if (i < end) {
    const int s0 = src_sorted[i];
    const float4 v0 = *(const float4*)(g + (size_t)s0 * H + lane * 4);
    acc0.x += v0.x; acc0.y += v0.y; acc0.z += v0.z; acc0.w += v0.w;
  }
  acc0.x += acc1.x; acc0.y += acc1.y; acc0.z += acc1.z; acc0.w += acc1.w;
  *(float4*)(agg + (size_t)p * H + lane * 4) = acc0;
}

// ---------------------------------------------------------------------------
// Final FC: out[M,8] = A[M,128] @ W[128,8] + b. Tiny-N, memory-bound -> VALU.
// ---------------------------------------------------------------------------
__global__ __launch_bounds__(256) void fc_out8(
    const float* __restrict__ A,
    const float* __restrict__ W,
    const float* __restrict__ b,
    float* __restrict__ C, int M)
{
  const int idx = blockIdx.x * 256 + threadIdx.x;
  if (idx >= M * OUT_D) return;
  const int i = idx >> 3;
  const int j = idx & 7;
  const float* ar = A + (size_t)i * H;
  float acc = b[j];
#pragma unroll 8
  for (int k = 0; k < H; ++k) acc = fmaf(ar[k], W[k * OUT_D + j], acc);
  C[idx] = acc;
}

// ---------------------------------------------------------------------------
extern "C" void kernel_launch(void* const* d_in, const int* in_sizes, int n_in,
                              void* d_out, int out_size, void* d_ws, size_t ws_size,
                              hipStream_t stream) {
  (void)in_sizes; (void)n_in; (void)out_size; (void)ws_size;

  const float* x_patient = (const float*)d_in[0];
  const float* x_gene    = (const float*)d_in[1];
  const int*   edges     = (const int*)d_in[2];
  const float* W_p1 = (const float*)d_in[3];
  const float* b_p1 = (const float*)d_in[4];
  const float* W_p2 = (const float*)d_in[5];
  const float* b_p2 = (const float*)d_in[6];
  const float* W_g1 = (const float*)d_in[7];
  const float* b_g1 = (const float*)d_in[8];
  const float* W_g2 = (const float*)d_in[9];
  const float* b_g2 = (const float*)d_in[10];
  const float* W1_l = (const float*)d_in[11];
  const float* b1_l = (const float*)d_in[12];
  const float* W1_r = (const float*)d_in[13];
  const float* W2_l = (const float*)d_in[14];
  const float* b2_l = (const float*)d_in[15];
  const float* W2_r = (const float*)d_in[16];
  const float* W_fc = (const float*)d_in[17];
  const float* b_fc = (const float*)d_in[18];
  float* out = (float*)d_out;

  // Workspace carve-up. hbuf is reused as p1; p is reused as p2.
  float* ws   = (float*)d_ws;
  float* g    = ws;                              // [N_GENE, H]
  float* hbuf = g    + (size_t)N_GENE * H;       // [N_PATIENT, H] (hidden / p1)
  float* p    = hbuf + (size_t)N_PATIENT * H;    // [N_PATIENT, H] (p / p2)
  float* agg  = p    + (size_t)N_PATIENT * H;    // [N_PATIENT, H]
  int* deg        = (int*)(agg + (size_t)N_PATIENT * H);   // [N_PATIENT]
  int* offs       = deg + N_PATIENT;                       // [N_PATIENT + 1]
  int* cursor     = offs + N_PATIENT + 1;                  // [N_PATIENT]
  int* src_sorted = cursor + N_PATIENT;                    // [N_EDGE]
  float* p1 = hbuf;
  float* p2 = p;

  const dim3 blk(256);
  const int gbG = (N_GENE + 63) / 64;
  const int gbP = (N_PATIENT + 63) / 64;
  const int gbE = (N_EDGE + 255) / 256;

  // --- CSR build (overlaps with the MLP GEMMs in issue order) ---
  hipMemsetAsync(deg, 0, (size_t)N_PATIENT * sizeof(int), stream);
  edge_count<<<gbE, blk, 0, stream>>>(edges, deg);
  deg_scan<<<1, SCAN_THREADS, 0, stream>>>(deg, offs, cursor);
  edge_fill<<<gbE, blk, 0, stream>>>(edges, cursor, src_sorted);

  // --- gene MLP ---
  gemm_h128<true,  false><<<gbG, blk, 0, stream>>>(x_gene, G_DIM, W_g1, nullptr, nullptr, b_g1, hbuf, N_GENE);
  gemm_h128<false, false><<<gbG, blk, 0, stream>>>(hbuf,   H,     W_g2, nullptr, nullptr, b_g2, g,    N_GENE);
  // --- patient MLP ---
  gemm_h128<true,  false><<<gbP, blk, 0, stream>>>(x_patient, P_DIM, W_p1, nullptr, nullptr, b_p1, hbuf, N_PATIENT);
  gemm_h128<false, false><<<gbP, blk, 0, stream>>>(hbuf,      H,     W_p2, nullptr, nullptr, b_p2, p,    N_PATIENT);
  // --- segment-sum as CSR gather ---
  gather_agg<<<(N_PATIENT + 7) / 8, blk, 0, stream>>>(g, offs, src_sorted, agg);
  // --- SAGE layers: relu(agg@W_l + prev@W_r + b) ---
  gemm_h128<true, true><<<gbP, blk, 0, stream>>>(agg, H, W1_l, p,  W1_r, b1_l, p1, N_PATIENT);
  gemm_h128<true, true><<<gbP, blk, 0, stream>>>(agg, H, W2_l, p1, W2_r, b2_l, p2, N_PATIENT);
  // --- final FC ---
  fc_out8<<<(N_PATIENT * OUT_D + 255) / 256, blk, 0, stream>>>(p2, W_fc, b_fc, out, N_PATIENT);
}